// HierarchicalGNN_16913581211654
// MI455X (gfx1250) — compile-verified
//
#include <hip/hip_runtime.h>
#include <hip/hip_bf16.h>
#include <math.h>

#define NN     4096
#define F_IN   128
#define OUTC   256
#define NHEAD  4
#define DHEAD  64
#define KNN    20
#define R_SH2  36.0f
#define R_MD2  100.0f
#define CAP_GAT 1024
#define CAP_GCN 512

typedef __attribute__((ext_vector_type(2))) float v2f;
typedef __attribute__((ext_vector_type(8))) float v8f;

__device__ __forceinline__ float elu_f(float v) { return v > 0.0f ? v : (__expf(v) - 1.0f); }

// ---------------------------------------------------------------------------
// Kernel 1: per-node top-20 nearest neighbors (excl self) + GCN degree norm.
// One thread per node; pos[j] loads are wave-uniform (scalarized, L2 resident).
// ---------------------------------------------------------------------------
__global__ void __launch_bounds__(256) topk_deg_kernel(const float* __restrict__ pos,
                                                       int* __restrict__ knn_idx,
                                                       float* __restrict__ nrm) {
    int n = blockIdx.x * 256 + threadIdx.x;
    if (n >= NN) return;
    float px = pos[n * 3 + 0], py = pos[n * 3 + 1], pz = pos[n * 3 + 2];
    float bd[KNN];
    int bi[KNN];
#pragma unroll
    for (int k = 0; k < KNN; ++k) { bd[k] = 3.4e38f; bi[k] = 0; }
    int deg = 0;
    for (int j = 0; j < NN; ++j) {
        if (j == n) continue;
        float dx = px - pos[j * 3 + 0];
        float dy = py - pos[j * 3 + 1];
        float dz = pz - pos[j * 3 + 2];
        float d2 = dx * dx + dy * dy + dz * dz;
        if (d2 <= R_SH2) deg++;
        if (d2 < bd[KNN - 1]) {
            float cd = d2; int ci = j;
#pragma unroll
            for (int k = 0; k < KNN; ++k) {
                if (cd < bd[k]) {
                    float td = bd[k]; int ti = bi[k];
                    bd[k] = cd; bi[k] = ci;
                    cd = td; ci = ti;
                }
            }
        }
    }
    nrm[n] = rsqrtf((float)(deg + 1));
#pragma unroll
    for (int k = 0; k < KNN; ++k) knn_idx[n * KNN + k] = bi[k];
}

// ---------------------------------------------------------------------------
// Kernel 2: f32 WMMA GEMM  C[M,N] = act(A*B (+ A2*B2) (+ bias))
// One wave per 16x16 C tile, V_WMMA_F32_16X16X4_F32, K-loop step 4.
// Fragment layouts per CDNA5 ISA 7.12.2 (half = lane>=16):
//   A 16x4 : a[r] = A[tileM + lane%16][k0 + 2*half + r]
//   B 4x16 : b[r] = B[k0 + 2*half + r][tileN + lane%16]
//   C 16x16: c[r] -> row tileM + r + 8*half, col tileN + lane%16
// ---------------------------------------------------------------------------
__global__ void __launch_bounds__(128) gemm_wmma_f32(const float* __restrict__ A,
                                                     const float* __restrict__ B,
                                                     const float* __restrict__ A2,
                                                     const float* __restrict__ B2,
                                                     const float* __restrict__ bias,
                                                     float* __restrict__ C,
                                                     int M, int Nc, int K, int act) {
    const int lane  = threadIdx.x;
    const int wave  = threadIdx.y;
    const int tileN = blockIdx.x * 16;
    const int tileM = (blockIdx.y * blockDim.y + wave) * 16;
    if (tileM >= M) return;           // uniform per wave -> EXEC stays all-ones
    const int half = lane >> 4;
    const int lm   = lane & 15;

    v8f acc = {};
    const float* arow = A + (size_t)(tileM + lm) * K;
    for (int k0 = 0; k0 < K; k0 += 4) {
        int ka = k0 + 2 * half;
        v2f a, b;
        a.x = arow[ka];
        a.y = arow[ka + 1];
        b.x = B[(size_t)ka * Nc + tileN + lm];
        b.y = B[(size_t)(ka + 1) * Nc + tileN + lm];
        acc = __builtin_amdgcn_wmma_f32_16x16x4_f32(false, a, false, b,
                                                    (short)0, acc, false, false);
    }
    if (A2) {
        const float* arow2 = A2 + (size_t)(tileM + lm) * K;
        for (int k0 = 0; k0 < K; k0 += 4) {
            int ka = k0 + 2 * half;
            v2f a, b;
            a.x = arow2[ka];
            a.y = arow2[ka + 1];
            b.x = B2[(size_t)ka * Nc + tileN + lm];
            b.y = B2[(size_t)(ka + 1) * Nc + tileN + lm];
            acc = __builtin_amdgcn_wmma_f32_16x16x4_f32(false, a, false, b,
                                                        (short)0, acc, false, false);
        }
    }
    float bv = bias ? bias[tileN + lm] : 0.0f;
#pragma unroll
    for (int r = 0; r < 8; ++r) {
        float v = acc[r] + bv;
        if (act) v = elu_f(v);
        C[(size_t)(tileM + r + 8 * half) * Nc + tileN + lm] = v;
    }
}

// ---------------------------------------------------------------------------
// Kernel 3: GAT attention scalars  a_s[n,h], a_d[n,h]
// ---------------------------------------------------------------------------
__global__ void __launch_bounds__(256) attvec_kernel(const float* __restrict__ xg,
                                                     const float* __restrict__ att_src,
                                                     const float* __restrict__ att_dst,
                                                     float* __restrict__ a_s,
                                                     float* __restrict__ a_d) {
    int t = blockIdx.x * 256 + threadIdx.x;
    if (t >= NN * NHEAD) return;
    int n = t >> 2, h = t & 3;
    const float* v = xg + (size_t)n * OUTC + h * DHEAD;
    float ss = 0.0f, sd = 0.0f;
#pragma unroll 4
    for (int d = 0; d < DHEAD; ++d) {
        float xv = v[d];
        ss += xv * att_src[h * DHEAD + d];
        sd += xv * att_dst[h * DHEAD + d];
    }
    a_s[t] = ss;
    a_d[t] = sd;
}

// ---------------------------------------------------------------------------
// Kernel 4: sparse GCN + GAT aggregation, one block per destination node.
// Deterministic LDS edge-list compaction (count -> serial scan -> ordered
// fill, no atomics), two-pass softmax, channel-parallel accumulation.
// ---------------------------------------------------------------------------
__global__ void __launch_bounds__(128) agg_kernel(const float* __restrict__ pos,
                                                  const float* __restrict__ a_s,
                                                  const float* __restrict__ a_d,
                                                  const float* __restrict__ nrm,
                                                  const float* __restrict__ xg,
                                                  const float* __restrict__ xw,
                                                  const float* __restrict__ b_gat,
                                                  const float* __restrict__ b_gcn,
                                                  float* __restrict__ z1,
                                                  float* __restrict__ z2) {
    __shared__ int   sj_gat[CAP_GAT];
    __shared__ float sw_gat[CAP_GAT * NHEAD];
    __shared__ int   sj_gcn[CAP_GCN];
    __shared__ float sw_gcn[CAP_GCN];
    __shared__ int   cnt_g[128], cnt_c[128];
    __shared__ int   off_g[129], off_c[129];
    __shared__ float red[128 * NHEAD];

    const int i = blockIdx.x;
    const int t = threadIdx.x;
    const float px = pos[i * 3 + 0], py = pos[i * 3 + 1], pz = pos[i * 3 + 2];
    float adi[NHEAD];
#pragma unroll
    for (int h = 0; h < NHEAD; ++h) adi[h] = a_d[i * NHEAD + h];

    // pass 1: count matches per thread (GAT mask reduces to d2<=100, diag incl.)
    int cg = 0, cc = 0;
    for (int j = t; j < NN; j += 128) {
        float dx = px - pos[j * 3 + 0];
        float dy = py - pos[j * 3 + 1];
        float dz = pz - pos[j * 3 + 2];
        float d2 = dx * dx + dy * dy + dz * dz;
        if (d2 <= R_MD2) cg++;
        if (d2 <= R_SH2 && j != i) cc++;
    }
    cnt_g[t] = cg; cnt_c[t] = cc;
    __syncthreads();
    if (t == 0) {
        int a = 0;
        for (int k = 0; k < 128; ++k) { off_g[k] = a; a += cnt_g[k]; }
        off_g[128] = a;
        int b = 1;                      // slot 0 reserved for GCN self loop
        for (int k = 0; k < 128; ++k) { off_c[k] = b; b += cnt_c[k]; }
        off_c[128] = b;
    }
    __syncthreads();

    // pass 2: ordered fill (deterministic order: by thread, then by j)
    int pg = off_g[t], pc = off_c[t];
    for (int j = t; j < NN; j += 128) {
        float dx = px - pos[j * 3 + 0];
        float dy = py - pos[j * 3 + 1];
        float dz = pz - pos[j * 3 + 2];
        float d2 = dx * dx + dy * dy + dz * dz;
        if (d2 <= R_MD2) {
            if (pg < CAP_GAT) {
                sj_gat[pg] = j;
#pragma unroll
                for (int h = 0; h < NHEAD; ++h) {
                    float l = adi[h] + a_s[j * NHEAD + h];
                    sw_gat[pg * NHEAD + h] = l > 0.0f ? l : 0.2f * l;   // leaky_relu
                }
            }
            pg++;
        }
        if (d2 <= R_SH2 && j != i) {
            if (pc < CAP_GCN) { sj_gcn[pc] = j; sw_gcn[pc] = nrm[j]; }
            pc++;
        }
    }
    if (t == 0) { sj_gcn[0] = i; sw_gcn[0] = nrm[i]; }
    __syncthreads();
    const int Eg = min(off_g[128], CAP_GAT);
    const int Ec = min(off_c[128], CAP_GCN);

    // softmax pass A: per-head max
    float m[NHEAD];
#pragma unroll
    for (int h = 0; h < NHEAD; ++h) m[h] = -3.4e38f;
    for (int e = t; e < Eg; e += 128)
#pragma unroll
        for (int h = 0; h < NHEAD; ++h) m[h] = fmaxf(m[h], sw_gat[e * NHEAD + h]);
#pragma unroll
    for (int h = 0; h < NHEAD; ++h) red[t * NHEAD + h] = m[h];
    __syncthreads();
    for (int s = 64; s > 0; s >>= 1) {
        if (t < s)
#pragma unroll
            for (int h = 0; h < NHEAD; ++h)
                red[t * NHEAD + h] = fmaxf(red[t * NHEAD + h], red[(t + s) * NHEAD + h]);
        __syncthreads();
    }
#pragma unroll
    for (int h = 0; h < NHEAD; ++h) m[h] = red[h];
    __syncthreads();

    // softmax pass B: exponentiate in place + per-head sum
    float sm[NHEAD] = {0.0f, 0.0f, 0.0f, 0.0f};
    for (int e = t; e < Eg; e += 128)
#pragma unroll
        for (int h = 0; h < NHEAD; ++h) {
            float w = __expf(sw_gat[e * NHEAD + h] - m[h]);
            sw_gat[e * NHEAD + h] = w;
            sm[h] += w;
        }
#pragma unroll
    for (int h = 0; h < NHEAD; ++h) red[t * NHEAD + h] = sm[h];
    __syncthreads();
    for (int s = 64; s > 0; s >>= 1) {
        if (t < s)
#pragma unroll
            for (int h = 0; h < NHEAD; ++h)
                red[t * NHEAD + h] += red[(t + s) * NHEAD + h];
        __syncthreads();
    }
    const int h0 = t >> 6, h1 = 2 + (t >> 6);
    const float inv_s0 = 1.0f / red[h0];
    const float inv_s1 = 1.0f / red[h1];
    __syncthreads();

    // channel-parallel accumulation: thread t owns channels t and t+128
    float ag0 = 0.0f, ag1 = 0.0f;
    for (int e = 0; e < Eg; ++e) {
        const float* xr = xg + (size_t)sj_gat[e] * OUTC;
        ag0 += sw_gat[e * NHEAD + h0] * xr[t];
        ag1 += sw_gat[e * NHEAD + h1] * xr[t + 128];
    }
    float ac0 = 0.0f, ac1 = 0.0f;
    for (int e = 0; e < Ec; ++e) {
        const float w = sw_gcn[e];
        const float* xr = xw + (size_t)sj_gcn[e] * OUTC;
        ac0 += w * xr[t];
        ac1 += w * xr[t + 128];
    }
    const float ni = nrm[i];
    z1[(size_t)i * OUTC + t]       = elu_f(ni * ac0 + b_gcn[t]);
    z1[(size_t)i * OUTC + t + 128] = elu_f(ni * ac1 + b_gcn[t + 128]);
    z2[(size_t)i * OUTC + t]       = elu_f(ag0 * inv_s0 + b_gat[t]);
    z2[(size_t)i * OUTC + t + 128] = elu_f(ag1 * inv_s1 + b_gat[t + 128]);
}

// ---------------------------------------------------------------------------
// Kernel 5: SAGE mean of K=20 neighbor features
// ---------------------------------------------------------------------------
__global__ void __launch_bounds__(128) sage_mean_kernel(const float* __restrict__ x,
                                                        const int* __restrict__ knn_idx,
                                                        float* __restrict__ mean_nb) {
    __shared__ int sidx[KNN];
    const int i = blockIdx.x, t = threadIdx.x;
    if (t < KNN) sidx[t] = knn_idx[i * KNN + t];
    __syncthreads();
    float s = 0.0f;
#pragma unroll
    for (int k = 0; k < KNN; ++k) s += x[(size_t)sidx[k] * F_IN + t];
    mean_nb[(size_t)i * F_IN + t] = s * (1.0f / (float)KNN);
}

// ---------------------------------------------------------------------------
// Kernel 6: gated fusion + layernorm, one 256-thread block per node
// ---------------------------------------------------------------------------
__global__ void __launch_bounds__(256) fuse_kernel(const float* __restrict__ z1,
                                                   const float* __restrict__ z2,
                                                   const float* __restrict__ z3,
                                                   const float* __restrict__ Wg,
                                                   const float* __restrict__ bg,
                                                   const float* __restrict__ gma,
                                                   const float* __restrict__ bta,
                                                   float* __restrict__ out) {
    __shared__ float r0[256], r1[256], r2[256];
    __shared__ float gsh[3];
    const int i = blockIdx.x, t = threadIdx.x;
    const float a = z1[(size_t)i * OUTC + t];
    const float b = z2[(size_t)i * OUTC + t];
    const float c = z3[(size_t)i * OUTC + t];
    r0[t] = a * Wg[t * 3 + 0] + b * Wg[(256 + t) * 3 + 0] + c * Wg[(512 + t) * 3 + 0];
    r1[t] = a * Wg[t * 3 + 1] + b * Wg[(256 + t) * 3 + 1] + c * Wg[(512 + t) * 3 + 1];
    r2[t] = a * Wg[t * 3 + 2] + b * Wg[(256 + t) * 3 + 2] + c * Wg[(512 + t) * 3 + 2];
    __syncthreads();
    for (int s = 128; s > 0; s >>= 1) {
        if (t < s) { r0[t] += r0[t + s]; r1[t] += r1[t + s]; r2[t] += r2[t + s]; }
        __syncthreads();
    }
    if (t == 0) {
        float l0 = r0[0] + bg[0], l1 = r1[0] + bg[1], l2 = r2[0] + bg[2];
        float mx = fmaxf(l0, fmaxf(l1, l2));
        float e0 = __expf(l0 - mx), e1 = __expf(l1 - mx), e2 = __expf(l2 - mx);
        float inv = 1.0f / (e0 + e1 + e2);
        gsh[0] = e0 * inv; gsh[1] = e1 * inv; gsh[2] = e2 * inv;
    }
    __syncthreads();
    const float zf = gsh[0] * a + gsh[1] * b + gsh[2] * c;
    r0[t] = zf; r1[t] = zf * zf;
    __syncthreads();
    for (int s = 128; s > 0; s >>= 1) {
        if (t < s) { r0[t] += r0[t + s]; r1[t] += r1[t + s]; }
        __syncthreads();
    }
    const float mu  = r0[0] * (1.0f / 256.0f);
    const float var = r1[0] * (1.0f / 256.0f) - mu * mu;
    out[(size_t)i * OUTC + t] = (zf - mu) * rsqrtf(var + 1e-5f) * gma[t] + bta[t];
}

// ---------------------------------------------------------------------------
extern "C" void kernel_launch(void* const* d_in, const int* in_sizes, int n_in,
                              void* d_out, int out_size, void* d_ws, size_t ws_size,
                              hipStream_t stream) {
    (void)in_sizes; (void)n_in; (void)out_size; (void)ws_size;
    const float* x       = (const float*)d_in[0];
    const float* pos     = (const float*)d_in[1];
    const float* W_gcn   = (const float*)d_in[2];
    const float* b_gcn   = (const float*)d_in[3];
    const float* W_gat   = (const float*)d_in[4];   // [128,4,64] == [128,256] row-major
    const float* att_src = (const float*)d_in[5];
    const float* att_dst = (const float*)d_in[6];
    const float* b_gat   = (const float*)d_in[7];
    const float* W_sage_l= (const float*)d_in[8];
    const float* W_sage_r= (const float*)d_in[9];
    const float* b_sage  = (const float*)d_in[10];
    const float* W_gate  = (const float*)d_in[11];
    const float* b_gate  = (const float*)d_in[12];
    const float* g_gamma = (const float*)d_in[13];
    const float* g_beta  = (const float*)d_in[14];

    float* outp = (float*)d_out;
    float* z1 = outp;
    float* z2 = outp + (size_t)NN * OUTC;
    float* z3 = outp + 2 * (size_t)NN * OUTC;
    float* fo = outp + 3 * (size_t)NN * OUTC;

    float* ws      = (float*)d_ws;
    float* xw      = ws;                         // [4096,256]
    float* xg      = xw + (size_t)NN * OUTC;     // [4096,256]
    float* a_s     = xg + (size_t)NN * OUTC;     // [4096,4]
    float* a_d     = a_s + (size_t)NN * NHEAD;   // [4096,4]
    float* nrm     = a_d + (size_t)NN * NHEAD;   // [4096]
    float* mean_nb = nrm + NN;                   // [4096,128]
    int*   knn_idx = (int*)(mean_nb + (size_t)NN * F_IN); // [4096,20]

    // 1. top-K neighbors + GCN degree norm
    topk_deg_kernel<<<NN / 256, 256, 0, stream>>>(pos, knn_idx, nrm);

    // 2. WMMA GEMMs: xw = x@W_gcn ; xg = x@W_gat
    dim3 gblk(32, 4);
    dim3 ggrd(OUTC / 16, NN / 64);
    gemm_wmma_f32<<<ggrd, gblk, 0, stream>>>(x, W_gcn, nullptr, nullptr, nullptr,
                                             xw, NN, OUTC, F_IN, 0);
    gemm_wmma_f32<<<ggrd, gblk, 0, stream>>>(x, W_gat, nullptr, nullptr, nullptr,
                                             xg, NN, OUTC, F_IN, 0);

    // 3. attention scalars
    attvec_kernel<<<(NN * NHEAD) / 256, 256, 0, stream>>>(xg, att_src, att_dst, a_s, a_d);

    // 4. sparse GCN + GAT aggregation -> z1, z2
    agg_kernel<<<NN, 128, 0, stream>>>(pos, a_s, a_d, nrm, xg, xw, b_gat, b_gcn, z1, z2);

    // 5. SAGE mean + fused dual-GEMM with bias + ELU -> z3
    sage_mean_kernel<<<NN, 128, 0, stream>>>(x, knn_idx, mean_nb);
    gemm_wmma_f32<<<ggrd, gblk, 0, stream>>>(mean_nb, W_sage_l, x, W_sage_r, b_sage,
                                             z3, NN, OUTC, F_IN, 1);

    // 6. gated fusion + layernorm -> out
    fuse_kernel<<<NN, 256, 0, stream>>>(z1, z2, z3, W_gate, b_gate, g_gamma, g_beta, fo);
}